// DropBlock_86517821213131
// MI455X (gfx1250) — compile-verified
//
#include <hip/hip_runtime.h>

// DropBlock, B=64 C=256 H=W=56, block_size=5, gamma=0.1.
// Bandwidth-bound: ~587MB of traffic -> ~25us roofline at 23.3TB/s.
// Dilation done entirely with 64-bit row bitmasks (2 wave32 ballots/row).
// u-planes staged into LDS via gfx1250 async-to-LDS DMA (ASYNCcnt path).
// x/out moved with b128 vector accesses (W=56 is float4-aligned per row).

#define PLANES   16384            // B*C
#define HB       52               // H - bs + 1
#define WB       52
#define HO       56
#define WO       56
#define UPLANE   (HB * WB)        // 2704 floats
#define XPLANE   (HO * WO)        // 3136 floats
#define XCHUNKS  (XPLANE / 4)     // 784 float4 per plane, 14 per row
#define COUNTM_F 51380224.0f      // 64*256*56*56
#define GAMMA_F  0.1f

typedef int v4i __attribute__((ext_vector_type(4)));
typedef v4i __attribute__((address_space(1))) * as1_v4i;
typedef v4i __attribute__((address_space(3))) * as3_v4i;

__device__ __forceinline__ void async_copy_b128(const float* g, float* l) {
#if __has_builtin(__builtin_amdgcn_global_load_async_to_lds_b128)
    __builtin_amdgcn_global_load_async_to_lds_b128(
        (as1_v4i)(unsigned long long)g,
        (as3_v4i)(unsigned)(unsigned long long)l,  // LDS offset = low 32 bits of flat addr
        0, 0);
#else
    unsigned lds = (unsigned)(unsigned long long)l;
    unsigned long long ga = (unsigned long long)g;
    asm volatile("global_load_async_to_lds_b128 %0, %1, off"
                 :: "v"(lds), "v"(ga) : "memory");
#endif
}

__device__ __forceinline__ void wait_async0() {
#if __has_builtin(__builtin_amdgcn_s_wait_asynccnt)
    __builtin_amdgcn_s_wait_asynccnt(0);
#else
    asm volatile("s_wait_asynccnt 0" ::: "memory");
#endif
}

__device__ __forceinline__ unsigned ballot32(bool p) {
#if __has_builtin(__builtin_amdgcn_ballot_w32)
    return __builtin_amdgcn_ballot_w32(p);
#else
    return (unsigned)__ballot(p);
#endif
}

// Stage one 52x52 u-plane into LDS (async DMA path), build per-row Bernoulli
// bitmasks with wave32 ballots, then produce the 56 dilated row masks.
// Requires blockDim.x == 256 (8 waves).
__device__ __forceinline__ void build_dilated(const float* __restrict__ uplane,
                                              float* us,
                                              unsigned long long* rowbits,
                                              unsigned long long* dil) {
    for (int i = threadIdx.x; i < UPLANE / 4; i += 256)
        async_copy_b128(uplane + i * 4, us + i * 4);
    wait_async0();          // this wave's async copies complete
    __syncthreads();        // all waves' copies visible in LDS

    const int wid  = threadIdx.x >> 5;
    const int lane = threadIdx.x & 31;
    for (int r = wid; r < HB; r += 8) {
        float a = us[r * WB + lane];
        unsigned b0 = ballot32(a < GAMMA_F);
        float b = (lane < (WB - 32)) ? us[r * WB + 32 + lane] : 1.0f;
        unsigned b1 = ballot32(b < GAMMA_F);
        if (lane == 0)
            rowbits[r] = (unsigned long long)b0 |
                         ((unsigned long long)(b1 & ((1u << (WB - 32)) - 1u)) << 32);
    }
    __syncthreads();

    if (threadIdx.x < HO) {
        int h  = threadIdx.x;
        int lo = (h - 4 < 0) ? 0 : h - 4;
        int hi = (h < HB - 1) ? h : HB - 1;
        unsigned long long m = 0ull;
        for (int r = lo; r <= hi; ++r) m |= rowbits[r];
        unsigned long long d = m | (m << 1) | (m << 2) | (m << 3) | (m << 4);
        dil[h] = d & ((1ull << WO) - 1ull);
    }
    __syncthreads();
}

__global__ void __launch_bounds__(256)
dropblock_count(const float* __restrict__ u,
                unsigned* __restrict__ cnt,
                unsigned long long* __restrict__ masks) {
    __shared__ __align__(16) float us[UPLANE];
    __shared__ unsigned long long rowbits[HB];
    __shared__ unsigned long long dil[HO];
    __shared__ unsigned s_sum;
    if (threadIdx.x == 0) s_sum = 0u;

    const int plane = blockIdx.x;
    build_dilated(u + (size_t)plane * UPLANE, us, rowbits, dil);

    if (threadIdx.x < HO) {
        unsigned long long d = dil[threadIdx.x];
        if (masks) masks[(size_t)plane * HO + threadIdx.x] = d;
        atomicAdd(&s_sum, (unsigned)(WO - __popcll(d)));
    }
    __syncthreads();
    if (threadIdx.x == 0) atomicAdd(cnt, s_sum);
}

// Vectorized apply: one float4 per iteration, b128 load/store. 14 chunks/row.
__device__ __forceinline__ void apply_plane(const float* __restrict__ xp,
                                            float* __restrict__ op,
                                            const unsigned long long* dil,
                                            float scale) {
    const float4* __restrict__ xp4 = (const float4*)xp;
    float4* __restrict__ op4       = (float4*)op;
    for (int i = threadIdx.x; i < XCHUNKS; i += 256) {
        int h  = i / 14;
        int w4 = (i - h * 14) * 4;
        float4 v = xp4[i];
        unsigned bits = (unsigned)(dil[h] >> w4) & 0xFu;
        float4 r;
        r.x = (bits & 1u) ? 0.0f : v.x * scale;
        r.y = (bits & 2u) ? 0.0f : v.y * scale;
        r.z = (bits & 4u) ? 0.0f : v.z * scale;
        r.w = (bits & 8u) ? 0.0f : v.w * scale;
        op4[i] = r;
    }
}

__global__ void __launch_bounds__(256)
dropblock_apply_masks(const float* __restrict__ x,
                      const unsigned long long* __restrict__ masks,
                      const unsigned* __restrict__ cnt,
                      float* __restrict__ out) {
    __shared__ unsigned long long dil[HO];
    const int plane = blockIdx.x;
    if (threadIdx.x < HO)
        dil[threadIdx.x] = masks[(size_t)plane * HO + threadIdx.x];
    __syncthreads();
    float scale = COUNTM_F / (float)(*cnt);
    apply_plane(x + (size_t)plane * XPLANE, out + (size_t)plane * XPLANE, dil, scale);
}

__global__ void __launch_bounds__(256)
dropblock_apply_recompute(const float* __restrict__ u,
                          const float* __restrict__ x,
                          const unsigned* __restrict__ cnt,
                          float* __restrict__ out) {
    __shared__ __align__(16) float us[UPLANE];
    __shared__ unsigned long long rowbits[HB];
    __shared__ unsigned long long dil[HO];
    const int plane = blockIdx.x;
    build_dilated(u + (size_t)plane * UPLANE, us, rowbits, dil);
    float scale = COUNTM_F / (float)(*cnt);
    apply_plane(x + (size_t)plane * XPLANE, out + (size_t)plane * XPLANE, dil, scale);
}

extern "C" void kernel_launch(void* const* d_in, const int* in_sizes, int n_in,
                              void* d_out, int out_size, void* d_ws, size_t ws_size,
                              hipStream_t stream) {
    (void)in_sizes; (void)n_in; (void)out_size;
    const float* x  = (const float*)d_in[0];
    const float* u  = (const float*)d_in[1];
    float* out      = (float*)d_out;

    unsigned* cnt = (unsigned*)d_ws;
    unsigned long long* masks = (unsigned long long*)((char*)d_ws + 16);
    const size_t need = 16 + (size_t)PLANES * HO * sizeof(unsigned long long);
    const bool use_masks = (ws_size >= need);

    (void)hipMemsetAsync(d_ws, 0, 16, stream);   // zero the survivor counter each call

    dropblock_count<<<PLANES, 256, 0, stream>>>(
        u, cnt, use_masks ? masks : (unsigned long long*)nullptr);

    if (use_masks)
        dropblock_apply_masks<<<PLANES, 256, 0, stream>>>(x, masks, cnt, out);
    else
        dropblock_apply_recompute<<<PLANES, 256, 0, stream>>>(u, x, cnt, out);
}